// DiagSSMBlock_335007449443
// MI455X (gfx1250) — compile-verified
//
#include <hip/hip_runtime.h>

// Problem sizes (match reference)
#define T_SEQ 4096
#define HID   1024
#define KDIM  1024

// GEMM tiling: block tile 128(M) x 64(N), 8 waves, each wave 32x32 (2x2 accs)
#define KC    32    // K chunk staged in LDS
#define BM    128
#define BN    64
#define LDX   36    // padded LDS row stride (floats): 16B aligned, conflict-light
#define LDB   36

// Scan chunking
#define CLEN   256
#define NCHUNK (T_SEQ / CLEN)

typedef float v2f __attribute__((ext_vector_type(2)));
typedef float v4f __attribute__((ext_vector_type(4)));
typedef float v8f __attribute__((ext_vector_type(8)));

// ---------------------------------------------------------------------------
// Kernel 1: S = X @ B  (fp32, V_WMMA_F32_16X16X4_F32), S written to d_out.
// X: (T_SEQ x KDIM) row-major, B: (KDIM x HID) row-major, S: (T_SEQ x HID).
// Each wave owns a 32x32 output tile => 4 accumulators, 1 LDS b64 per WMMA.
// ---------------------------------------------------------------------------
__global__ __launch_bounds__(256) void gemm_xb_wmma(const float* __restrict__ X,
                                                    const float* __restrict__ B,
                                                    float* __restrict__ S) {
  __shared__ float sX[BM * LDX];   // X tile, row-major [m][k]
  __shared__ float sB[BN * LDB];   // B tile, transposed [n][k]

  const int tid  = threadIdx.x;
  const int wave = tid >> 5;
  const int lane = tid & 31;
  const int half = lane >> 4;   // 0: lanes 0-15, 1: lanes 16-31
  const int r    = lane & 15;

  const int blockN = blockIdx.x * BN;
  const int blockM = blockIdx.y * BM;

  const int wm = (wave & 3) << 5;   // wave M offset within block tile (0,32,64,96)
  const int wn = (wave >> 2) << 5;  // wave N offset within block tile (0,32)

  // staging decomposition
  const int rowGrp = tid >> 3;          // 0..31
  const int col4   = (tid & 7) << 2;    // 0,4,...,28   (X staging, KC=32 wide)
  const int bRow   = tid >> 4;          // 0..15        (B staging helper)
  const int bCol4  = (tid & 15) << 2;   // 0,4,...,60   (BN=64 wide)

  v8f acc00 = {}, acc01 = {}, acc10 = {}, acc11 = {};

  for (int k0 = 0; k0 < KDIM; k0 += KC) {
    __syncthreads();  // previous chunk's fragment reads done before restaging

    // Stage X tile: BM x KC, coalesced float4 loads, row-major in LDS
    #pragma unroll
    for (int p = 0; p < 4; ++p) {
      const int row = rowGrp + (p << 5);            // 0..127
      v4f xv = *(const v4f*)(X + (size_t)(blockM + row) * KDIM + k0 + col4);
      *(v4f*)(&sX[row * LDX + col4]) = xv;
    }
    // Stage B tile transposed: KC rows x BN cols, write [n][k]
    #pragma unroll
    for (int p = 0; p < 2; ++p) {
      const int kRow = bRow + (p << 4);             // 0..31
      v4f bv = *(const v4f*)(B + (size_t)(k0 + kRow) * HID + blockN + bCol4);
      #pragma unroll
      for (int i = 0; i < 4; ++i)
        sB[(bCol4 + i) * LDB + kRow] = bv[i];
    }
    __syncthreads();

    // 8 K-steps x 4 WMMAs per chunk
    #pragma unroll
    for (int kk = 0; kk < KC; kk += 4) {
      // A frags: lane(half,r) = X[m][k0+kk+2*half], +1 (contiguous float2)
      v2f a0 = *(const v2f*)(&sX[(wm + r)      * LDX + kk + 2 * half]);
      v2f a1 = *(const v2f*)(&sX[(wm + 16 + r) * LDX + kk + 2 * half]);
      // B frags: lane(half,r) = B[k0+kk+2*half][n], +1 (contiguous in sB^T)
      v2f b0 = *(const v2f*)(&sB[(wn + r)      * LDB + kk + 2 * half]);
      v2f b1 = *(const v2f*)(&sB[(wn + 16 + r) * LDB + kk + 2 * half]);
      acc00 = __builtin_amdgcn_wmma_f32_16x16x4_f32(false, a0, false, b0,
                                                    (short)0, acc00, false, false);
      acc01 = __builtin_amdgcn_wmma_f32_16x16x4_f32(false, a0, false, b1,
                                                    (short)0, acc01, false, false);
      acc10 = __builtin_amdgcn_wmma_f32_16x16x4_f32(false, a1, false, b0,
                                                    (short)0, acc10, false, false);
      acc11 = __builtin_amdgcn_wmma_f32_16x16x4_f32(false, a1, false, b1,
                                                    (short)0, acc11, false, false);
    }
  }

  // D layout per 16x16 tile: VGPR i -> row (i + 8*half), col r
  #pragma unroll
  for (int i = 0; i < 8; ++i) {
    const int row0 = blockM + wm + i + 8 * half;
    const int col0 = blockN + wn + r;
    S[(size_t)row0        * HID + col0]      = acc00[i];
    S[(size_t)row0        * HID + col0 + 16] = acc01[i];
    S[(size_t)(row0 + 16) * HID + col0]      = acc10[i];
    S[(size_t)(row0 + 16) * HID + col0 + 16] = acc11[i];
  }
}

// ---------------------------------------------------------------------------
// Kernel 2: per-chunk local scan, in place on d_out.
// h_t = a*h_{t-1} + s_t within each chunk (zero initial state);
// final chunk state written to carry[c][h].
// ---------------------------------------------------------------------------
__global__ __launch_bounds__(256) void scan_local(const float* __restrict__ a_diag,
                                                  float* __restrict__ out,
                                                  float* __restrict__ carry) {
  const int h = blockIdx.x * 256 + threadIdx.x;   // channel
  const int c = blockIdx.y;                       // chunk
  const float a = a_diag[h];
  float accv = 0.0f;
  const int t0 = c * CLEN;
  for (int t = 0; t < CLEN; ++t) {
    const size_t idx = (size_t)(t0 + t) * HID + h;
    accv = fmaf(a, accv, out[idx]);
    out[idx] = accv;
  }
  carry[c * HID + h] = accv;
}

// ---------------------------------------------------------------------------
// Kernel 3: exclusive prefix over chunk carries with decay a^CLEN.
// carryIn[c] = state entering chunk c.
// ---------------------------------------------------------------------------
__global__ __launch_bounds__(256) void scan_carry(const float* __restrict__ a_diag,
                                                  const float* __restrict__ carry,
                                                  float* __restrict__ carryIn) {
  const int h = blockIdx.x * 256 + threadIdx.x;
  const float a = a_diag[h];
  float aL = a;                       // a^CLEN via 8 squarings (CLEN = 256)
  #pragma unroll
  for (int i = 0; i < 8; ++i) aL *= aL;
  float state = 0.0f;
  for (int c = 0; c < NCHUNK; ++c) {
    carryIn[c * HID + h] = state;
    state = fmaf(aL, state, carry[c * HID + h]);
  }
}

// ---------------------------------------------------------------------------
// Kernel 4: apply cross-chunk correction: out[t] += a^(tlocal+1) * carryIn[c].
// Chunk 0 needs no correction -> grid.y covers chunks 1..NCHUNK-1.
// ---------------------------------------------------------------------------
__global__ __launch_bounds__(256) void scan_fix(const float* __restrict__ a_diag,
                                                const float* __restrict__ carryIn,
                                                float* __restrict__ out) {
  const int h = blockIdx.x * 256 + threadIdx.x;
  const int c = blockIdx.y + 1;
  const float a  = a_diag[h];
  const float ci = carryIn[c * HID + h];
  float w = a;
  const int t0 = c * CLEN;
  for (int t = 0; t < CLEN; ++t) {
    const size_t idx = (size_t)(t0 + t) * HID + h;
    out[idx] = fmaf(w, ci, out[idx]);
    w *= a;
  }
}

// ---------------------------------------------------------------------------
extern "C" void kernel_launch(void* const* d_in, const int* in_sizes, int n_in,
                              void* d_out, int out_size, void* d_ws, size_t ws_size,
                              hipStream_t stream) {
  const float* x_seq  = (const float*)d_in[0];  // (T, H) fp32
  const float* a_diag = (const float*)d_in[1];  // (H,)   fp32
  const float* b_mat  = (const float*)d_in[2];  // (H, H) fp32
  float* out = (float*)d_out;                   // (T, H) fp32

  float* carry   = (float*)d_ws;                // NCHUNK * H floats
  float* carryIn = carry + NCHUNK * HID;        // NCHUNK * H floats (128 KB total)

  // 1) S = X @ B  -> d_out
  dim3 gGemm(HID / BN, T_SEQ / BM);             // (16, 32) blocks
  gemm_xb_wmma<<<gGemm, 256, 0, stream>>>(x_seq, b_mat, out);

  // 2) chunked causal scan, in place
  dim3 gLocal(HID / 256, NCHUNK);               // (4, 16)
  scan_local<<<gLocal, 256, 0, stream>>>(a_diag, out, carry);

  scan_carry<<<HID / 256, 256, 0, stream>>>(a_diag, carry, carryIn);

  dim3 gFix(HID / 256, NCHUNK - 1);             // (4, 15)
  scan_fix<<<gFix, 256, 0, stream>>>(a_diag, carryIn, out);
}